// GNFConverter_2010044695218
// MI455X (gfx1250) — compile-verified
//
#include <hip/hip_runtime.h>
#include <math.h>

typedef __attribute__((ext_vector_type(2))) float v2f;
typedef __attribute__((ext_vector_type(8))) float v8f;

#define NB 16
#define NA 128
#define NP 8192
#define NT 5
#define CLIPT 0.3f
#define WAVES_PER_BLOCK 8
#define QPB (WAVES_PER_BLOCK * 16) /* 128 queries per block */

__global__ __launch_bounds__(256) void gnf_softmax_field_kernel(
    const float* __restrict__ coords,     // [B, A, 3]
    const int*   __restrict__ atom_types, // [B, A]
    const float* __restrict__ query,      // [B, P, 3]
    float* __restrict__ out)              // [B, P, T, 3]
{
    __shared__ float sx[NA], sy[NA], sz[NA], sn2[NA];
    __shared__ int   st[NA];

    const int b     = blockIdx.y;
    const int qbase = blockIdx.x * QPB;
    const int tid   = threadIdx.x;

    // Stage atom data (coords, |c|^2, type) for this batch into LDS once.
    if (tid < NA) {
        const float x = coords[(b * NA + tid) * 3 + 0];
        const float y = coords[(b * NA + tid) * 3 + 1];
        const float z = coords[(b * NA + tid) * 3 + 2];
        sx[tid]  = x;
        sy[tid]  = y;
        sz[tid]  = z;
        sn2[tid] = x * x + y * y + z * z;
        st[tid]  = atom_types[b * NA + tid];
    }
    __syncthreads();

    const int lane = tid & 31;
    const int wave = tid >> 5;
    const int half = lane >> 4;   // 0 = lanes 0-15, 1 = lanes 16-31
    const int l16  = lane & 15;

    // This lane mirrors query q (duplicated across the two lane-halves).
    const int   q   = qbase + wave * 16 + l16;
    const float qx  = query[(b * NP + q) * 3 + 0];
    const float qy  = query[(b * NP + q) * 3 + 1];
    const float qz  = query[(b * NP + q) * 3 + 2];
    const float qn2 = qx * qx + qy * qy + qz * qz;

    // B operand: 4x16 f32, K rows = query coords (x,y,z,0).
    // Low lanes carry K=0 (V0) and K=1 (V1); high lanes K=2 (V0) and K=3 (V1).
    v2f Bop;
    Bop[0] = half ? qz : qx;
    Bop[1] = half ? 0.0f : qy;

    // Per-type accumulators: sum(e), sum(e*coord). All compile-time indexed.
    float accS[NT], accX[NT], accY[NT], accZ[NT];
#pragma unroll
    for (int t = 0; t < NT; ++t) { accS[t] = 0.f; accX[t] = 0.f; accY[t] = 0.f; accZ[t] = 0.f; }

#pragma unroll
    for (int tile = 0; tile < NA / 16; ++tile) {
        const int am = tile * 16 + l16;
        // A operand: 16x4 f32, rows = atoms, pre-scaled by -2 so the WMMA
        // directly accumulates -2 * (c . q).
        const float a_lo0 = sx[am];
        const float a_lo1 = sy[am];
        const float a_hi0 = sz[am];
        v2f Aop;
        Aop[0] = -2.0f * (half ? a_hi0 : a_lo0);
        Aop[1] = half ? 0.0f : (-2.0f * a_lo1);

        v8f c = {};
        // D[m][n] = -2 * dot(coord_m, query_n) for this 16-atom tile.
        v8f d = __builtin_amdgcn_wmma_f32_16x16x4_f32(
            /*neg_a=*/false, Aop, /*neg_b=*/false, Bop,
            /*c_mod=*/(short)0, c, /*reuse_a=*/false, /*reuse_b=*/false);

        // C/D layout: VGPR r, lane l -> atom (tile*16 + r + 8*half), query l%16.
#pragma unroll
        for (int r = 0; r < 8; ++r) {
            const int   a  = tile * 16 + r + 8 * half;
            const float cx = sx[a];
            const float cy = sy[a];
            const float cz = sz[a];
            float dist2 = sn2[a] + qn2 + d[r];
            dist2 = fmaxf(dist2, 0.0f);
            // logits <= 0 always: plain exp is numerically safe, no max-shift.
            const float e  = __expf(-__builtin_amdgcn_sqrtf(dist2));
            const float ex = e * cx;
            const float ey = e * cy;
            const float ez = e * cz;
            const int   ty = st[a];
            // Branchless per-type scatter: 0/1 mask + FMA (no EXEC churn).
#pragma unroll
            for (int t = 0; t < NT; ++t) {
                const float m = (ty == t) ? 1.0f : 0.0f;
                accS[t] = fmaf(m, e,  accS[t]);
                accX[t] = fmaf(m, ex, accX[t]);
                accY[t] = fmaf(m, ey, accY[t]);
                accZ[t] = fmaf(m, ez, accZ[t]);
            }
        }
    }

    // Merge the two lane-halves (each saw a disjoint half of the 128 atoms
    // for the same query).
#pragma unroll
    for (int t = 0; t < NT; ++t) {
        accS[t] += __shfl_xor(accS[t], 16, 32);
        accX[t] += __shfl_xor(accX[t], 16, 32);
        accY[t] += __shfl_xor(accY[t], 16, 32);
        accZ[t] += __shfl_xor(accZ[t], 16, 32);
    }

    if (half == 0) {
        float* o = out + (size_t)(b * NP + q) * (NT * 3);
#pragma unroll
        for (int t = 0; t < NT; ++t) {
            const float S = accS[t];
            float gx, gy, gz;
            if (S > 0.0f) {
                const float inv = __builtin_amdgcn_rcpf(S);
                gx = fmaf(accX[t], inv, -qx);   // weighted mean coord - query
                gy = fmaf(accY[t], inv, -qy);
                gz = fmaf(accZ[t], inv, -qz);
            } else {                             // type with no atoms -> zero field
                gx = 0.f; gy = 0.f; gz = 0.f;
            }
            const float mag = __builtin_amdgcn_sqrtf(gx * gx + gy * gy + gz * gz);
            const float s   = CLIPT * __builtin_amdgcn_rcpf(fmaxf(mag, CLIPT));
            o[t * 3 + 0] = gx * s;
            o[t * 3 + 1] = gy * s;
            o[t * 3 + 2] = gz * s;
        }
    }
}

extern "C" void kernel_launch(void* const* d_in, const int* in_sizes, int n_in,
                              void* d_out, int out_size, void* d_ws, size_t ws_size,
                              hipStream_t stream) {
    (void)in_sizes; (void)n_in; (void)d_ws; (void)ws_size; (void)out_size;
    const float* coords     = (const float*)d_in[0]; // [16,128,3]
    const int*   atom_types = (const int*)  d_in[1]; // [16,128]
    const float* query      = (const float*)d_in[2]; // [16,8192,3]
    float*       out        = (float*)d_out;         // [16,8192,5,3]

    dim3 grid(NP / QPB, NB);  // (64, 16)
    dim3 block(256);          // 8 wave32s
    gnf_softmax_field_kernel<<<grid, block, 0, stream>>>(coords, atom_types, query, out);
}